// CrossPlaneMixer_60636348285068
// MI455X (gfx1250) — compile-verified
//
#include <hip/hip_runtime.h>

typedef __attribute__((ext_vector_type(2))) float v2f;
typedef __attribute__((ext_vector_type(8))) float v8f;

#define SPATIAL    65536      // 64*64*16 per (b,c) plane
#define POOL_ELEMS 131072     // B*C*64*16 per pooled table

// ---------------------------------------------------------------------------
// Stage 1: build the 6 pooled-mean tables (each entry = mean of 64 inputs).
//   t=0 P_dh_H [B,C,D=64,Hp=16]   t=1 P_dh_D [B,C,Dp=16,H=64]
//   t=2 P_dw_W [B,C,D=64,Wp=16]   t=3 P_dw_D [B,C,Dp=16,W=64]
//   t=4 P_hw_W [B,C,H=64,Wp=16]   t=5 P_hw_H [B,C,Hp=16,W=64]
// ---------------------------------------------------------------------------
__global__ __launch_bounds__(256) void pool6_kernel(
    const float* __restrict__ dh, const float* __restrict__ dw,
    const float* __restrict__ hw, float* __restrict__ ws)
{
    const int t = blockIdx.y;
    const int i = blockIdx.x * blockDim.x + threadIdx.x;   // 0..131071
    const int bc = i >> 10;
    const int r  = i & 1023;
    const int base = bc * SPATIAL;
    float s = 0.f;

    if (t == 0) {                       // P_dh_H: r = d*16 + hp  (64 contiguous)
        const int d = r >> 4, hp = r & 15;
        const float* p = dh + base + d * 1024 + hp * 64;
        #pragma unroll
        for (int j = 0; j < 64; ++j) s += p[j];
    } else if (t == 1) {                // P_dh_D: r = dp*64 + h
        const int dp = r >> 6, h = r & 63;
        const float* p = dh + base + dp * 4096 + h * 16;
        #pragma unroll
        for (int j = 0; j < 4; ++j)
            #pragma unroll
            for (int w = 0; w < 16; ++w) s += p[j * 1024 + w];
    } else if (t == 2) {                // P_dw_W: r = d*16 + wp
        const int d = r >> 4, wp = r & 15;
        const float* p = dw + base + d * 1024 + wp * 4;
        #pragma unroll
        for (int h = 0; h < 16; ++h)
            #pragma unroll
            for (int j = 0; j < 4; ++j) s += p[h * 64 + j];
    } else if (t == 3) {                // P_dw_D: r = dp*64 + w
        const int dp = r >> 6, w = r & 63;
        const float* p = dw + base + dp * 4096 + w;
        #pragma unroll
        for (int j = 0; j < 4; ++j)
            #pragma unroll
            for (int h = 0; h < 16; ++h) s += p[j * 1024 + h * 64];
    } else if (t == 4) {                // P_hw_W: r = h*16 + wp
        const int h = r >> 4, wp = r & 15;
        const float* p = hw + base + h * 64 + wp * 4;
        #pragma unroll
        for (int d = 0; d < 16; ++d)
            #pragma unroll
            for (int j = 0; j < 4; ++j) s += p[d * 4096 + j];
    } else {                            // P_hw_H: r = hp*64 + w
        const int hp = r >> 6, w = r & 63;
        const float* p = hw + base + hp * 256 + w;
        #pragma unroll
        for (int d = 0; d < 16; ++d)
            #pragma unroll
            for (int j = 0; j < 4; ++j) s += p[d * 4096 + j * 64];
    }
    ws[t * POOL_ELEMS + i] = s * (1.0f / 64.0f);
}

// ---------------------------------------------------------------------------
// Stage 2: fused concat + 1x1x1 conv as an M=64, K=192, N=16-per-wave GEMM
// using V_WMMA_F32_16X16X4_F32. One wave = one 64x16 output tile.
//   A (16x4)  : weights from LDS; lane l: row = m0+(l&15), cols k+2g, k+2g+1
//   B (4x16)  : activations;      lane l: col = l&15, rows 2g, 2g+1
//   C/D (16x16): VGPR r -> M = r (lanes 0-15) / r+8 (lanes 16-31)
// ---------------------------------------------------------------------------
__global__ __launch_bounds__(256) void mix_gemm_kernel(
    const float* __restrict__ x0,   // direct input plane  (channel stride 65536)
    const float* __restrict__ p1,   // pooled table 1      (channel stride 1024)
    const float* __restrict__ p2,   // pooled table 2      (channel stride 1024)
    const float* __restrict__ wgt,  // [64][192] row-major
    const float* __restrict__ bias, // [64]
    float* __restrict__ out,        // this plane's output
    int plane)
{
    __shared__ float lds_w[64 * 192];
    __shared__ float lds_b[64];
    {
        const float4* src = (const float4*)wgt;
        float4*       dst = (float4*)lds_w;
        for (int i = threadIdx.x; i < (64 * 192) / 4; i += blockDim.x) dst[i] = src[i];
        if (threadIdx.x < 64) lds_b[threadIdx.x] = bias[threadIdx.x];
    }
    __syncthreads();

    const int wave = threadIdx.x >> 5;
    const int lane = threadIdx.x & 31;
    const int tile = blockIdx.x * (blockDim.x >> 5) + wave;   // 0..8191
    const int n  = lane & 15;
    const int g  = lane >> 4;
    const int r0 = g * 2;            // B-matrix rows (K) held by this lane

    int b, d, h, w0, s0, off1, off2, ns1, ns2;
    if (plane == 0) {                // dh: D=64 H=64 W=16
        b = tile >> 12; d = (tile >> 6) & 63; h = tile & 63; w0 = 0;
        s0   = d * 1024 + h * 16;
        off1 = d * 16;  ns1 = 1;     // P_dw_W[d, w]
        off2 = h * 16;  ns2 = 1;     // P_hw_W[h, w]
    } else if (plane == 1) {         // dw: D=64 H=16 W=64
        b = tile >> 12; d = (tile >> 6) & 63; h = (tile >> 2) & 15; w0 = (tile & 3) * 16;
        s0   = d * 1024 + h * 64 + w0;
        off1 = d * 16 + h;       ns1 = 0;   // P_dh_H[d, h] broadcast over w
        off2 = h * 64 + w0;      ns2 = 1;   // P_hw_H[h, w]
    } else {                         // hw: D=16 H=64 W=64
        b = tile >> 12; d = (tile >> 8) & 15; h = (tile >> 2) & 63; w0 = (tile & 3) * 16;
        s0   = d * 4096 + h * 64 + w0;
        off1 = d * 64 + h;       ns1 = 0;   // P_dh_D[d, h] broadcast over w
        off2 = d * 64 + w0;      ns2 = 1;   // P_dw_D[d, w]
    }

    const int bc0 = b * 64;
    const float* bp0 = x0 + bc0 * SPATIAL + s0 + n;
    const float* bp1 = p1 + bc0 * 1024 + off1 + n * ns1;
    const float* bp2 = p2 + bc0 * 1024 + off2 + n * ns2;

    v8f acc[4];
    #pragma unroll
    for (int mt = 0; mt < 4; ++mt)
        #pragma unroll
        for (int r = 0; r < 8; ++r)
            acc[mt][r] = lds_b[mt * 16 + r + 8 * g];

    // ---- section 0: direct input, channel stride = SPATIAL, weight cols 0..63
    #pragma unroll
    for (int kk = 0; kk < 64; kk += 4) {
        v2f bf;
        bf.x = bp0[(kk + r0)     * SPATIAL];
        bf.y = bp0[(kk + r0 + 1) * SPATIAL];
        #pragma unroll
        for (int mt = 0; mt < 4; ++mt) {
            v2f af = *(const v2f*)(lds_w + (mt * 16 + n) * 192 + kk + r0);
            acc[mt] = __builtin_amdgcn_wmma_f32_16x16x4_f32(
                false, af, false, bf, (short)0, acc[mt], false, false);
        }
    }
    // ---- section 1: pooled table 1, channel stride = 1024, weight cols 64..127
    #pragma unroll
    for (int kk = 0; kk < 64; kk += 4) {
        v2f bf;
        bf.x = bp1[(kk + r0)     * 1024];
        bf.y = bp1[(kk + r0 + 1) * 1024];
        #pragma unroll
        for (int mt = 0; mt < 4; ++mt) {
            v2f af = *(const v2f*)(lds_w + (mt * 16 + n) * 192 + 64 + kk + r0);
            acc[mt] = __builtin_amdgcn_wmma_f32_16x16x4_f32(
                false, af, false, bf, (short)0, acc[mt], false, false);
        }
    }
    // ---- section 2: pooled table 2, channel stride = 1024, weight cols 128..191
    #pragma unroll
    for (int kk = 0; kk < 64; kk += 4) {
        v2f bf;
        bf.x = bp2[(kk + r0)     * 1024];
        bf.y = bp2[(kk + r0 + 1) * 1024];
        #pragma unroll
        for (int mt = 0; mt < 4; ++mt) {
            v2f af = *(const v2f*)(lds_w + (mt * 16 + n) * 192 + 128 + kk + r0);
            acc[mt] = __builtin_amdgcn_wmma_f32_16x16x4_f32(
                false, af, false, bf, (short)0, acc[mt], false, false);
        }
    }

    // ---- store: D[M][N], M = mt*16 + r + 8g, N = n
    float* op = out + bc0 * SPATIAL + s0 + n;
    #pragma unroll
    for (int mt = 0; mt < 4; ++mt)
        #pragma unroll
        for (int r = 0; r < 8; ++r)
            op[(mt * 16 + r + 8 * g) * SPATIAL] = acc[mt][r];
}

// ---------------------------------------------------------------------------
extern "C" void kernel_launch(void* const* d_in, const int* in_sizes, int n_in,
                              void* d_out, int out_size, void* d_ws, size_t ws_size,
                              hipStream_t stream) {
    const float* dh   = (const float*)d_in[0];
    const float* dw   = (const float*)d_in[1];
    const float* hw   = (const float*)d_in[2];
    const float* w_dh = (const float*)d_in[3];
    const float* b_dh = (const float*)d_in[4];
    const float* w_dw = (const float*)d_in[5];
    const float* b_dw = (const float*)d_in[6];
    const float* w_hw = (const float*)d_in[7];
    const float* b_hw = (const float*)d_in[8];
    float* out = (float*)d_out;
    float* ws  = (float*)d_ws;

    // Stage 1: pooled tables (3 MB workspace)
    pool6_kernel<<<dim3(POOL_ELEMS / 256, 6), dim3(256), 0, stream>>>(dh, dw, hw, ws);

    float* P_dh_H = ws + 0 * POOL_ELEMS;
    float* P_dh_D = ws + 1 * POOL_ELEMS;
    float* P_dw_W = ws + 2 * POOL_ELEMS;
    float* P_dw_D = ws + 3 * POOL_ELEMS;
    float* P_hw_W = ws + 4 * POOL_ELEMS;
    float* P_hw_H = ws + 5 * POOL_ELEMS;

    // Stage 2: three WMMA GEMMs (8192 tiles each, 8 waves/block)
    mix_gemm_kernel<<<dim3(1024), dim3(256), 0, stream>>>(
        dh, P_dw_W, P_hw_W, w_dh, b_dh, out + 0,        0);
    mix_gemm_kernel<<<dim3(1024), dim3(256), 0, stream>>>(
        dw, P_dh_H, P_hw_H, w_dw, b_dw, out + 8388608,  1);
    mix_gemm_kernel<<<dim3(1024), dim3(256), 0, stream>>>(
        hw, P_dh_D, P_dw_D, w_hw, b_hw, out + 16777216, 2);
}